// ModelBasicGNN_54889682043540
// MI455X (gfx1250) — compile-verified
//
#include <hip/hip_runtime.h>

typedef __attribute__((ext_vector_type(2))) float v2f;
typedef __attribute__((ext_vector_type(8))) float v8f;

#define NEG_SLOPE 0.1f

__device__ __forceinline__ float lrelu(float x) { return x >= 0.f ? x : NEG_SLOPE * x; }

__device__ __forceinline__ void atomic_add_f32(float* p, float v) {
    __hip_atomic_fetch_add(p, v, __ATOMIC_RELAXED, __HIP_MEMORY_SCOPE_AGENT);
}

// ---------------------------------------------------------------------------
// Small utility kernels
// ---------------------------------------------------------------------------
__global__ void fill_kernel(float* __restrict__ p, float v, int n) {
    int i = blockIdx.x * blockDim.x + threadIdx.x;
    if (i < n) p[i] = v;
}

// deg[dst[e]] += ew[e]   (deg pre-filled with 1.0 for the self loop)
__global__ void deg_edge_kernel(const int* __restrict__ dst, const float* __restrict__ ew,
                                float* __restrict__ deg, int E) {
    int e = blockIdx.x * blockDim.x + threadIdx.x;
    if (e < E) atomic_add_f32(&deg[dst[e]], ew[e]);
}

// in-place deg -> dinv = rsqrt(deg)
__global__ void dinv_kernel(float* __restrict__ d, int n) {
    int i = blockIdx.x * blockDim.x + threadIdx.x;
    if (i < n) {
        float v = d[i];
        d[i] = v > 0.f ? rsqrtf(v) : 0.f;
    }
}

// norm[e] = dinv[src]*ew*dinv[dst]  (identical across all 3 layers -> compute once)
__global__ void norm_kernel(const int* __restrict__ src, const int* __restrict__ dst,
                            const float* __restrict__ ew, const float* __restrict__ dinv,
                            float* __restrict__ norm, int E) {
    int e = blockIdx.x * blockDim.x + threadIdx.x;
    if (e < E) norm[e] = dinv[src[e]] * ew[e] * dinv[dst[e]];
}

// ---------------------------------------------------------------------------
// f32 WMMA GEMM:  Y[M,N] = act(X[M,K]) @ W[K,N] (+ bias) (opt. leaky out)
// One wave computes one 16x16 tile of Y with V_WMMA_F32_16X16X4_F32,
// fully unrolled over K (compile-time, K % 4 == 0).
// Requirements (hold for this model): K % 4 == 0, M % 16 == 0.
// ---------------------------------------------------------------------------
#define GEMM_WAVES 4

template <int K, int LIN, int LOUT>
__global__ void gemm_wmma_kernel(const float* __restrict__ X, const float* __restrict__ W,
                                 const float* __restrict__ bias, float* __restrict__ Y,
                                 int M, int N) {
    const int lane  = threadIdx.x & 31;
    const int wave  = threadIdx.x >> 5;
    const int mbase = (blockIdx.x * GEMM_WAVES + wave) * 16;
    const int nbase = blockIdx.y * 16;
    if (mbase >= M) return;               // uniform per-wave exit

    const int  half = lane >> 4;          // 0: lanes 0-15, 1: lanes 16-31
    const int  lr   = lane & 15;
    const int  m    = mbase + lr;         // A row for this lane (always < M)
    const int  n    = nbase + lr;         // B/C column for this lane
    const bool nok  = (n < N);
    const int  ncl  = nok ? n : (N - 1);  // clamped column: branch-free B loads

    v8f acc = {0.f, 0.f, 0.f, 0.f, 0.f, 0.f, 0.f, 0.f};

    // ISA 7.12.2: A 16x4 f32: lanes0-15 v0=K0 v1=K1; lanes16-31 v0=K2 v1=K3
    const float* xrow = X + (long)m * K + 2 * half;

    #pragma unroll
    for (int k0 = 0; k0 < K; k0 += 4) {
        v2f a = *reinterpret_cast<const v2f*>(xrow + k0);   // b64 load, 8B aligned
        if (LIN) { a.x = lrelu(a.x); a.y = lrelu(a.y); }
        const int ka0 = k0 + 2 * half;
        float b0 = W[ka0 * N + ncl];
        float b1 = W[(ka0 + 1) * N + ncl];
        v2f b;
        b.x = nok ? b0 : 0.f;                               // v_cndmask, no branch
        b.y = nok ? b1 : 0.f;
        acc = __builtin_amdgcn_wmma_f32_16x16x4_f32(false, a, false, b,
                                                    (short)0, acc, false, false);
    }

    const float bv = (bias != nullptr) ? bias[ncl] : 0.f;
    if (nok) {
        #pragma unroll
        for (int r = 0; r < 8; ++r) {
            // C/D layout: VGPR r -> row mbase+r (lanes 0-15) / mbase+r+8 (lanes 16-31)
            const int mm = mbase + r + 8 * half;
            float v = acc[r] + bv;
            if (LOUT) v = lrelu(v);
            Y[(long)mm * N + n] = v;
        }
    }
}

// ---------------------------------------------------------------------------
// agg[v][j] = bias[j] + dinv[v]^2 * xw[v][j]        (self-loop + bias fused)
// ---------------------------------------------------------------------------
__global__ void agg_init_kernel(const float* __restrict__ xw, const float* __restrict__ dinv,
                                const float* __restrict__ bias, float* __restrict__ agg,
                                int total, int dout) {
    int t = blockIdx.x * blockDim.x + threadIdx.x;
    if (t < total) {
        int v = t / dout;
        int j = t - v * dout;
        float di = dinv[v];
        agg[t] = bias[j] + di * di * xw[t];
    }
}

// ---------------------------------------------------------------------------
// Edge aggregation: agg[dst] += norm[e] * xw[src]   (float4 per thread)
// thread t -> (edge = t/dout4, chunk = t%dout4) so adjacent lanes touch
// contiguous bytes of the same gathered row (coalesced; L2 resident).
// ---------------------------------------------------------------------------
__global__ void edge_agg_kernel(const int* __restrict__ src, const int* __restrict__ dst,
                                const float* __restrict__ nrm, const float* __restrict__ xw,
                                float* __restrict__ agg, int total, int dout4, int dout) {
    int t = blockIdx.x * blockDim.x + threadIdx.x;
    if (t >= total) return;
    int e = t / dout4;
    int c = t - e * dout4;
    int s = src[e];
    int d = dst[e];
    float w = nrm[e];
    float4 v = reinterpret_cast<const float4*>(xw + (long)s * dout)[c];
    float* ar = agg + (long)d * dout + 4 * c;
    atomic_add_f32(ar + 0, w * v.x);
    atomic_add_f32(ar + 1, w * v.y);
    atomic_add_f32(ar + 2, w * v.z);
    atomic_add_f32(ar + 3, w * v.w);
}

// ---------------------------------------------------------------------------
// Sum-pool per graph with fused leaky-ReLU on the node activations.
// ---------------------------------------------------------------------------
__global__ void pool_kernel(const float* __restrict__ h, const int* __restrict__ batch,
                            float* __restrict__ pooled, int total, int dout4, int dout) {
    int t = blockIdx.x * blockDim.x + threadIdx.x;
    if (t >= total) return;
    int v = t / dout4;
    int c = t - v * dout4;
    int g = batch[v];
    float4 x = reinterpret_cast<const float4*>(h + (long)v * dout)[c];
    float* pr = pooled + (long)g * dout + 4 * c;
    atomic_add_f32(pr + 0, lrelu(x.x));
    atomic_add_f32(pr + 1, lrelu(x.y));
    atomic_add_f32(pr + 2, lrelu(x.z));
    atomic_add_f32(pr + 3, lrelu(x.w));
}

// ---------------------------------------------------------------------------
// Launch
// ---------------------------------------------------------------------------
static inline dim3 gemm_grid(int M, int N) {
    int mtiles = (M + 15) / 16;
    return dim3((mtiles + GEMM_WAVES - 1) / GEMM_WAVES, (N + 15) / 16);
}

extern "C" void kernel_launch(void* const* d_in, const int* in_sizes, int n_in,
                              void* d_out, int out_size, void* d_ws, size_t ws_size,
                              hipStream_t stream) {
    const float* x0    = (const float*)d_in[0];
    const int*   eidx  = (const int*)d_in[1];
    const float* ew    = (const float*)d_in[2];
    const int*   batch = (const int*)d_in[3];
    const float* W1 = (const float*)d_in[4];   const float* b1  = (const float*)d_in[5];
    const float* W2 = (const float*)d_in[6];   const float* b2  = (const float*)d_in[7];
    const float* W3 = (const float*)d_in[8];   const float* b3  = (const float*)d_in[9];
    const float* Wf1 = (const float*)d_in[10]; const float* bf1 = (const float*)d_in[11];
    const float* Wf2 = (const float*)d_in[12]; const float* bf2 = (const float*)d_in[13];
    const float* Wf3 = (const float*)d_in[14]; const float* bf3 = (const float*)d_in[15];

    const int N = in_sizes[0] / 4;   // nodes (4 input features)
    const int E = in_sizes[1] / 2;   // edges
    const int G = out_size / 2;      // graphs (2 output features)
    const int* src = eidx;
    const int* dst = eidx + E;

    // workspace carving (256B aligned)
    char*  ws  = (char*)d_ws;
    size_t off = 0;
    auto carve = [&](size_t bytes) -> float* {
        float* p = (float*)(ws + off);
        off = (off + bytes + 255) & ~(size_t)255;
        return p;
    };
    float* dinv   = carve((size_t)N * 4);            // degrees -> dinv (in place)
    float* norm   = carve((size_t)E * 4);            // per-edge norm (shared by all layers)
    float* bufA   = carve((size_t)N * 48 * 4);       // xw
    float* bufB   = carve((size_t)N * 48 * 4);       // agg / h
    float* pooled = carve((size_t)G * 48 * 4);
    float* t1     = carve((size_t)G * 32 * 4);
    float* t2     = carve((size_t)G * 16 * 4);
    (void)ws_size; (void)n_in;

    const int B = 256;
    auto blocks = [&](int n) { return (n + B - 1) / B; };

    // ---- degree / normalization (computed once, reused by all 3 layers) ----
    fill_kernel<<<blocks(N), B, 0, stream>>>(dinv, 1.0f, N);                  // self loop
    deg_edge_kernel<<<blocks(E), B, 0, stream>>>(dst, ew, dinv, E);
    dinv_kernel<<<blocks(N), B, 0, stream>>>(dinv, N);
    norm_kernel<<<blocks(E), B, 0, stream>>>(src, dst, ew, dinv, norm, E);

    auto gcn_post = [&](const float* bias, int dout) {
        agg_init_kernel<<<blocks(N * dout), B, 0, stream>>>(bufA, dinv, bias, bufB, N * dout, dout);
        int d4 = dout / 4;
        edge_agg_kernel<<<blocks(E * d4), B, 0, stream>>>(src, dst, norm, bufA, bufB, E * d4, d4, dout);
    };

    // ---- 3 GCN layers (activation of prev layer fused into next gemm load) ----
    gemm_wmma_kernel< 4, 0, 0><<<gemm_grid(N, 12), 32 * GEMM_WAVES, 0, stream>>>(x0,   W1, nullptr, bufA, N, 12);
    gcn_post(b1, 12);
    gemm_wmma_kernel<12, 1, 0><<<gemm_grid(N, 24), 32 * GEMM_WAVES, 0, stream>>>(bufB, W2, nullptr, bufA, N, 24);
    gcn_post(b2, 24);
    gemm_wmma_kernel<24, 1, 0><<<gemm_grid(N, 48), 32 * GEMM_WAVES, 0, stream>>>(bufB, W3, nullptr, bufA, N, 48);
    gcn_post(b3, 48);

    // ---- sum-pool per graph (leaky fused into load) ----
    fill_kernel<<<blocks(G * 48), B, 0, stream>>>(pooled, 0.0f, G * 48);
    pool_kernel<<<blocks(N * 12), B, 0, stream>>>(bufB, batch, pooled, N * 12, 12, 48);

    // ---- MLP head ----
    gemm_wmma_kernel<48, 0, 1><<<gemm_grid(G, 32), 32 * GEMM_WAVES, 0, stream>>>(pooled, Wf1, bf1, t1, G, 32);
    gemm_wmma_kernel<32, 0, 1><<<gemm_grid(G, 16), 32 * GEMM_WAVES, 0, stream>>>(t1,     Wf2, bf2, t2, G, 16);
    gemm_wmma_kernel<16, 0, 0><<<gemm_grid(G,  2), 32 * GEMM_WAVES, 0, stream>>>(t2,     Wf3, bf3, (float*)d_out, G, 2);
}